// HRP_14516989460570
// MI455X (gfx1250) — compile-verified
//
#include <hip/hip_runtime.h>

typedef __attribute__((ext_vector_type(2))) float v2f;
typedef __attribute__((ext_vector_type(8))) float v8f;

#define BB 16384
#define TT 2
#define NN 100
#define SS 6
#define HH 16

__device__ __forceinline__ v8f wmma4(v2f a, v2f b, v8f c) {
  // D = A(16x4,f32) * B(4x16,f32) + C(16x16,f32)
  return __builtin_amdgcn_wmma_f32_16x16x4_f32(false, a, false, b, (short)0, c, false, false);
}

__device__ __forceinline__ v8f splat8(float v) {
  v8f r;
#pragma unroll
  for (int i = 0; i < 8; ++i) r[i] = v;
  return r;
}

#define LOG2E 1.4426950408889634f
// Hardware transcendental paths: v_exp_f32 + v_rcp_f32 (TRANS32, co-executes with WMMA)
__device__ __forceinline__ float fast_exp(float x) { return __builtin_amdgcn_exp2f(x * LOG2E); }
__device__ __forceinline__ float sigmoidf(float v) {
  return __builtin_amdgcn_rcpf(1.0f + fast_exp(-v));   // ±inf saturate correctly
}
__device__ __forceinline__ float tanh_fast(float v) {
  return 1.0f - 2.0f * __builtin_amdgcn_rcpf(1.0f + fast_exp(2.0f * v));
}

// -------------------- Kernel A: per-n GRU + attention MLP via f32 WMMA --------------------
// One wave handles a tile of 16 batch rows for one n.
// Writes a[b,t,n] to Aout in (n,b,t) flat order == reference p_flat order.
__global__ __launch_bounds__(128) void fused_a(
    const float* __restrict__ x,
    const float* __restrict__ Wih, const float* __restrict__ Whh,
    const float* __restrict__ bih, const float* __restrict__ bhh,
    const float* __restrict__ Wa1, const float* __restrict__ ba1,
    const float* __restrict__ Wa2, const float* __restrict__ ba2,
    const float* __restrict__ Wa3, const float* __restrict__ ba3,
    float* __restrict__ Aout)
{
  __shared__ __align__(16) float smem[4 * 256];
  const int tid  = threadIdx.x;
  const int wv   = tid >> 5;
  const int lane = tid & 31;
  const int hf   = lane >> 4;   // lane half (0/1)
  const int l16  = lane & 15;
  float* sm = smem + wv * 256;

  const int pair = blockIdx.x * 4 + wv;          // 0 .. 102399
  const int n    = pair % NN;
  const int b0   = (pair / NN) * 16;
  const int brow = b0 + l16;

  // ---- x as A-tiles: ax[t][c], element (M=l16, K=4c+2*hf+v), K==s, zero for s>=6 ----
  v2f ax[TT][2];
#pragma unroll
  for (int t = 0; t < TT; ++t) {
    const float* xp = x + (((size_t)brow * TT + t) * NN + n) * SS;
    float2 lo = *(const float2*)(xp + 2 * hf);
    ax[t][0].x = lo.x; ax[t][0].y = lo.y;
    if (hf == 0) { float2 hi2 = *(const float2*)(xp + 4); ax[t][1].x = hi2.x; ax[t][1].y = hi2.y; }
    else         { ax[t][1].x = 0.f; ax[t][1].y = 0.f; }
  }

  // ---- B tiles (K x 16): element (K=4c+2*hf+v, Ncol=l16) ----
  v2f Bih[3][2];                       // W_ih^T : B[s][k] = Wih[n][k][s], s>=6 -> 0
#pragma unroll
  for (int ct = 0; ct < 3; ++ct) {
    const float* wp = Wih + ((size_t)n * 48 + 16 * ct + l16) * SS;
    float2 v0 = *(const float2*)(wp + 2 * hf);
    Bih[ct][0].x = v0.x; Bih[ct][0].y = v0.y;
    if (hf == 0) { float2 v1 = *(const float2*)(wp + 4); Bih[ct][1].x = v1.x; Bih[ct][1].y = v1.y; }
    else         { Bih[ct][1].x = 0.f; Bih[ct][1].y = 0.f; }
  }
  v2f Bhh[3][4];                       // W_hh^T
#pragma unroll
  for (int ct = 0; ct < 3; ++ct)
#pragma unroll
    for (int c = 0; c < 4; ++c) {
      float2 v = *(const float2*)(Whh + ((size_t)n * 48 + 16 * ct + l16) * HH + 4 * c + 2 * hf);
      Bhh[ct][c].x = v.x; Bhh[ct][c].y = v.y;
    }
  v2f Bw1[4], Bw2[4];                  // Wa1^T, Wa2^T
#pragma unroll
  for (int c = 0; c < 4; ++c) {
    float2 v1 = *(const float2*)(Wa1 + ((size_t)n * HH + l16) * HH + 4 * c + 2 * hf);
    Bw1[c].x = v1.x; Bw1[c].y = v1.y;
    float2 v2 = *(const float2*)(Wa2 + ((size_t)n * HH + l16) * HH + 4 * c + 2 * hf);
    Bw2[c].x = v2.x; Bw2[c].y = v2.y;
  }
  const float bi0 = bih[n * 48 + l16], bi1 = bih[n * 48 + 16 + l16], bi2 = bih[n * 48 + 32 + l16];
  const float bh0 = bhh[n * 48 + l16], bh1 = bhh[n * 48 + 16 + l16], bh2 = bhh[n * 48 + 32 + l16];
  const float ba1v = ba1[n * HH + l16], ba2v = ba2[n * HH + l16], ba3v = ba3[n];
  float4 w3[4];
#pragma unroll
  for (int c = 0; c < 4; ++c) w3[c] = *(const float4*)(Wa3 + n * HH + 4 * c);

  // ---- gi = x @ W_ih^T + b_ih  (includes input bias), C-layout tiles ----
  v8f GI[TT][3];
#pragma unroll
  for (int t = 0; t < TT; ++t)
#pragma unroll
    for (int ct = 0; ct < 3; ++ct) {
      v8f c = splat8(ct == 0 ? bi0 : (ct == 1 ? bi1 : bi2));
      c = wmma4(ax[t][0], Bih[ct][0], c);
      c = wmma4(ax[t][1], Bih[ct][1], c);
      GI[t][ct] = c;
    }

  // ---- GRU t=0 (h=0 -> gh = b_hh) ----
  v8f h1;
#pragma unroll
  for (int e = 0; e < 8; ++e) {
    float r  = sigmoidf(GI[0][0][e] + bh0);
    float z  = sigmoidf(GI[0][1][e] + bh1);
    float nn = tanh_fast(GI[0][2][e] + r * bh2);
    h1[e] = (1.f - z) * nn;
  }

  // C-layout (M over vgprs, N over lanes) -> A-layout (M over lanes, K over vgpr/half) via LDS
  auto xpose = [&](const v8f& tile, v2f out[4]) {
    __syncthreads();
#pragma unroll
    for (int e = 0; e < 8; ++e) sm[(e + 8 * hf) * 16 + l16] = tile[e];
    __syncthreads();
#pragma unroll
    for (int c = 0; c < 4; ++c) {
      float2 v = *(const float2*)(sm + l16 * 16 + 4 * c + 2 * hf);
      out[c].x = v.x; out[c].y = v.y;
    }
  };

  // ---- GRU t=1: gh = h1 @ W_hh^T + b_hh ----
  v2f ah1[4]; xpose(h1, ah1);
  v8f GH0 = splat8(bh0), GH1 = splat8(bh1), GH2 = splat8(bh2);
#pragma unroll
  for (int c = 0; c < 4; ++c) {
    GH0 = wmma4(ah1[c], Bhh[0][c], GH0);
    GH1 = wmma4(ah1[c], Bhh[1][c], GH1);
    GH2 = wmma4(ah1[c], Bhh[2][c], GH2);
  }
  v8f h2;
#pragma unroll
  for (int e = 0; e < 8; ++e) {
    float r  = sigmoidf(GI[1][0][e] + GH0[e]);
    float z  = sigmoidf(GI[1][1][e] + GH1[e]);
    float nn = tanh_fast(GI[1][2][e] + r * GH2[e]);
    h2[e] = (1.f - z) * nn + z * h1[e];
  }
  v2f ah2[4]; xpose(h2, ah2);

  // ---- attention MLP per t ----
#pragma unroll
  for (int t = 0; t < TT; ++t) {
    v2f* A0 = (t == 0) ? ah1 : ah2;
    v8f a1 = splat8(ba1v);
#pragma unroll
    for (int c = 0; c < 4; ++c) a1 = wmma4(A0[c], Bw1[c], a1);
#pragma unroll
    for (int e = 0; e < 8; ++e) a1[e] = fmaxf(a1[e], 0.f);
    v2f at1[4]; xpose(a1, at1);
    v8f a2 = splat8(ba2v);
#pragma unroll
    for (int c = 0; c < 4; ++c) a2 = wmma4(at1[c], Bw2[c], a2);
    __syncthreads();
#pragma unroll
    for (int e = 0; e < 8; ++e) sm[(e + 8 * hf) * 16 + l16] = fmaxf(a2[e], 0.f);
    __syncthreads();
    if (lane < 16) {
      const float4* rp = (const float4*)(sm + l16 * 16);
      float dot = ba3v;
#pragma unroll
      for (int c = 0; c < 4; ++c) {
        float4 r4 = rp[c];
        dot += r4.x * w3[c].x + r4.y * w3[c].y + r4.z * w3[c].z + r4.w * w3[c].w;
      }
      // a[b,t,n] stored at flat (n,b,t) == p_flat order
      Aout[(size_t)n * (BB * TT) + (size_t)brow * TT + t] = fmaxf(dot, 0.f);
    }
  }
}

// -------------------- Kernel C: p[:, -1] gather (contiguous in p_flat order) --------------------
__global__ __launch_bounds__(256) void copy_p_last(const float* __restrict__ A, float* __restrict__ out)
{
  int idx = blockIdx.x * 256 + threadIdx.x;   // b'*100 + n'
  if (idx < BB * NN) {
    int b = idx / NN, nn = idx % NN;
    out[idx] = A[(size_t)b * (TT * NN) + (TT - 1) * NN + nn];
  }
}

// -------------------- Kernel B: xi build + f-MLP + GRU-c + q --------------------
__global__ __launch_bounds__(32) void fused_b(
    const float* __restrict__ x, const float* __restrict__ A,
    const float* __restrict__ Wl1, const float* __restrict__ bl1,
    const float* __restrict__ Wl2, const float* __restrict__ bl2,
    const float* __restrict__ Wl3, const float* __restrict__ bl3,
    const float* __restrict__ Wihc, const float* __restrict__ bihc,
    const float* __restrict__ Whhc, const float* __restrict__ bhhc,
    const float* __restrict__ Wc1, const float* __restrict__ bc1,
    const float* __restrict__ Wc2, const float* __restrict__ bc2,
    const float* __restrict__ Wc3, const float* __restrict__ bc3,
    float* __restrict__ outq)
{
  // weight cache offsets
  constexpr int OW1 = 0, OB1 = 1225, OW2 = 1260, OB2 = 2485, OW3 = 2520, OB3 = 2555,
                OWI = 2556, OBI = 4236, OWH = 4284, OBH = 5052,
                OC1 = 5100, OD1 = 5356, OC2 = 5372, OD2 = 5628, OC3 = 5644, OD3 = 5660;
  __shared__ float sW[5661];
  __shared__ float sP[32 * 177];    // per-thread vectors (LDS to avoid scratch), odd stride: no bank conflicts
  const int tid = threadIdx.x;

#define LOADW(off, src, cnt) for (int i = tid; i < (cnt); i += 32) sW[(off) + i] = (src)[i]
  LOADW(OW1, Wl1, 1225); LOADW(OB1, bl1, 35);
  LOADW(OW2, Wl2, 1225); LOADW(OB2, bl2, 35);
  LOADW(OW3, Wl3, 35);   LOADW(OB3, bl3, 1);
  LOADW(OWI, Wihc, 1680); LOADW(OBI, bihc, 48);
  LOADW(OWH, Whhc, 768);  LOADW(OBH, bhhc, 48);
  LOADW(OC1, Wc1, 256);  LOADW(OD1, bc1, 16);
  LOADW(OC2, Wc2, 256);  LOADW(OD2, bc2, 16);
  LOADW(OC3, Wc3, 16);   LOADW(OD3, bc3, 1);
#undef LOADW
  __syncthreads();

  float* P   = sP + tid * 177;
  float* xi  = P;            // [2][35]
  float* bufA = P + 70;      // [35]
  float* bufB = P + 105;     // [35]
  float* hc  = P + 140;      // [16]
  float* hn  = P + 156;      // [16]

  const int b = blockIdx.x * 32 + tid;

  // ---- build xi[t][35] = [xc(:,:,99), xc(:,:,89), 0, 0, xc(:,:,98)] ----
  for (int t = 0; t < TT; ++t) {
    const float* xb = x + (size_t)(b * TT + t) * NN * SS;
    const float* Ab = A + (size_t)b * (TT * NN) + t * NN;   // p_view[b,t,:]
    float* xt = xi + t * 35;
#pragma unroll
    for (int j = 0; j < 6; ++j) {
      xt[j]      = xb[99 * SS + j];
      xt[7 + j]  = xb[89 * SS + j];
      xt[28 + j] = xb[98 * SS + j];
    }
    xt[6]  = Ab[99];
    xt[13] = Ab[89];
    xt[34] = Ab[98];
#pragma unroll
    for (int j = 14; j < 28; ++j) xt[j] = 0.f;
  }

  // ---- f = MLP3(xi[t=1]) ----
  for (int o = 0; o < 35; ++o) {
    float s = sW[OB1 + o];
    const float* w = sW + OW1 + o * 35;
    for (int j = 0; j < 35; ++j) s += w[j] * xi[35 + j];
    bufA[o] = fmaxf(s, 0.f);
  }
  for (int o = 0; o < 35; ++o) {
    float s = sW[OB2 + o];
    const float* w = sW + OW2 + o * 35;
    for (int j = 0; j < 35; ++j) s += w[j] * bufA[j];
    bufB[o] = fmaxf(s, 0.f);
  }
  float f = sW[OB3];
  for (int j = 0; j < 35; ++j) f += sW[OW3 + j] * bufB[j];
  f = fmaxf(f, 0.f);

  // ---- GRU-c over T=2 ----
  for (int k = 0; k < 16; ++k) hc[k] = 0.f;
  for (int t = 0; t < TT; ++t) {
    const float* xt = xi + t * 35;
    for (int k = 0; k < 16; ++k) {
      float ir = sW[OBI + k], iz = sW[OBI + 16 + k], inn = sW[OBI + 32 + k];
      for (int j = 0; j < 35; ++j) {
        float xv = xt[j];
        ir  += sW[OWI + k * 35 + j] * xv;
        iz  += sW[OWI + (16 + k) * 35 + j] * xv;
        inn += sW[OWI + (32 + k) * 35 + j] * xv;
      }
      float hr = sW[OBH + k], hz = sW[OBH + 16 + k], hnb = sW[OBH + 32 + k];
      for (int j = 0; j < 16; ++j) {
        float hv = hc[j];
        hr  += sW[OWH + k * 16 + j] * hv;
        hz  += sW[OWH + (16 + k) * 16 + j] * hv;
        hnb += sW[OWH + (32 + k) * 16 + j] * hv;
      }
      float r = sigmoidf(ir + hr), z = sigmoidf(iz + hz);
      float nn2 = tanh_fast(inn + r * hnb);
      hn[k] = (1.f - z) * nn2 + z * hc[k];
    }
    for (int k = 0; k < 16; ++k) hc[k] = hn[k];
  }

  // ---- qc = MLP3(hc) ----
  for (int o = 0; o < 16; ++o) {
    float s = sW[OD1 + o];
    for (int j = 0; j < 16; ++j) s += sW[OC1 + o * 16 + j] * hc[j];
    hn[o] = fmaxf(s, 0.f);
  }
  for (int o = 0; o < 16; ++o) {
    float s = sW[OD2 + o];
    for (int j = 0; j < 16; ++j) s += sW[OC2 + o * 16 + j] * hn[j];
    hc[o] = fmaxf(s, 0.f);
  }
  float qc = sW[OD3];
  for (int j = 0; j < 16; ++j) qc += sW[OC3 + j] * hc[j];
  qc = fmaxf(qc, 0.f);

  outq[b] = f + qc;
}

extern "C" void kernel_launch(void* const* d_in, const int* in_sizes, int n_in,
                              void* d_out, int out_size, void* d_ws, size_t ws_size,
                              hipStream_t stream) {
  (void)in_sizes; (void)n_in; (void)out_size; (void)ws_size;
  const float* x     = (const float*)d_in[0];
  const float* Wih   = (const float*)d_in[1];
  const float* Whh   = (const float*)d_in[2];
  const float* bih   = (const float*)d_in[3];
  const float* bhh   = (const float*)d_in[4];
  const float* Wa1   = (const float*)d_in[5];
  const float* ba1   = (const float*)d_in[6];
  const float* Wa2   = (const float*)d_in[7];
  const float* ba2   = (const float*)d_in[8];
  const float* Wa3   = (const float*)d_in[9];
  const float* ba3   = (const float*)d_in[10];
  const float* Wl1   = (const float*)d_in[11];
  const float* bl1   = (const float*)d_in[12];
  const float* Wl2   = (const float*)d_in[13];
  const float* bl2   = (const float*)d_in[14];
  const float* Wl3   = (const float*)d_in[15];
  const float* bl3   = (const float*)d_in[16];
  const float* Wihc  = (const float*)d_in[17];
  const float* Whhc  = (const float*)d_in[18];
  const float* bihc  = (const float*)d_in[19];
  const float* bhhc  = (const float*)d_in[20];
  const float* Wc1   = (const float*)d_in[21];
  const float* bc1   = (const float*)d_in[22];
  const float* Wc2   = (const float*)d_in[23];
  const float* bc2   = (const float*)d_in[24];
  const float* Wc3   = (const float*)d_in[25];
  const float* bc3   = (const float*)d_in[26];

  float* Abuf = (float*)d_ws;                 // N*B*T floats = 13.1 MB, (n,b,t) flat
  float* out  = (float*)d_out;                // [B*N p_last | B q]

  // (B/16)*N waves, 4 waves (128 threads) per block
  fused_a<<<(BB / 16) * NN / 4, 128, 0, stream>>>(x, Wih, Whh, bih, bhh,
                                                  Wa1, ba1, Wa2, ba2, Wa3, ba3, Abuf);
  copy_p_last<<<(BB * NN + 255) / 256, 256, 0, stream>>>(Abuf, out);
  fused_b<<<BB / 32, 32, 0, stream>>>(x, Abuf,
                                      Wl1, bl1, Wl2, bl2, Wl3, bl3,
                                      Wihc, bihc, Whhc, bhhc,
                                      Wc1, bc1, Wc2, bc2, Wc3, bc3,
                                      out + (size_t)BB * NN);
}